// KNNInteractionGraph_4260607557911
// MI455X (gfx1250) — compile-verified
//
#include <hip/hip_runtime.h>

#define N_ATOMS  8192
#define K_NB     32
#define CUTOFF_F 10.0f
#define NUM_MOLS 128

typedef __attribute__((ext_vector_type(2))) float v2f;
typedef __attribute__((ext_vector_type(8))) float v8f;

// ---------------------------------------------------------------------------
// Kernel 1: molecule segment starts via binary search on the sorted batch[].
// start[m] = first i with batch[i] >= m, for m = 0..NUM_MOLS (start[128]==N).
// ---------------------------------------------------------------------------
__global__ void mol_ranges_kernel(const int* __restrict__ batch,
                                  int* __restrict__ start,
                                  int n, int num_mols) {
  int m = blockIdx.x * blockDim.x + threadIdx.x;
  if (m > num_mols) return;
  int lo = 0, hi = n;
  while (lo < hi) {
    int mid = (lo + hi) >> 1;
    if (batch[mid] < m) lo = mid + 1; else hi = mid;
  }
  start[m] = lo;
}

// ---------------------------------------------------------------------------
// Kernel 2: one wave32 per 16-row tile. WMMA f32 16x16x4 gram tiles over the
// tile's molecule segment; per-row top-32 maintained in LDS.
// ---------------------------------------------------------------------------
__global__ __launch_bounds__(32)
void knn_tile_kernel(const float* __restrict__ pos,
                     const int*  __restrict__ batch,
                     const int*  __restrict__ mstart,
                     int*  __restrict__ eidx,   // [2, N*K] flat
                     float* __restrict__ ew) {  // [N*K]
  __shared__ float s_tile[16 * 16];     // distance tile (row-major M,N)
  __shared__ float s_rsq[16];           // row squared norms
  __shared__ int   s_rb[16];            // row batch ids
  __shared__ float s_lw[16][K_NB];      // per-row sorted top-K distances
  __shared__ int   s_li[16][K_NB];      // per-row top-K neighbor indices

  const int lane = threadIdx.x;   // 0..31
  const int half = lane >> 4;     // 0 or 1 (K / M-half split in WMMA layout)
  const int l16  = lane & 15;
  const int r0   = blockIdx.x * 16;

  // init per-row top-K lists
  for (int t = lane; t < 16 * K_NB; t += 32) {
    ((float*)s_lw)[t] = CUTOFF_F;
    ((int*)s_li)[t]   = 0;
  }

  // row position / norm / batch (lanes 0..15 publish to LDS)
  const int r = r0 + l16;
  const float px = pos[3 * r + 0];
  const float py = pos[3 * r + 1];
  const float pz = pos[3 * r + 2];
  if (half == 0) {
    s_rsq[l16] = px * px + py * py + pz * pz;
    s_rb[l16]  = batch[r];
  }
  __syncthreads();

  // A fragment: lane L holds row M=l16, K = {2*half, 2*half+1}; coords (x,y,z,0)
  v2f afrag;
  afrag.x = (half == 0) ? px : pz;
  afrag.y = (half == 0) ? py : 0.0f;

  // cache row norms/batches for this lane's 8 accumulator rows M = v + 8*half
  float rsq[8];
  int   rb[8];
#pragma unroll
  for (int v = 0; v < 8; ++v) {
    rsq[v] = s_rsq[v + 8 * half];
    rb[v]  = s_rb[v + 8 * half];
  }

  // uniform candidate column range = union of molecule segments of rows r0..r0+15
  const int b_lo    = batch[r0];
  const int b_hi    = batch[r0 + 15];
  const int c_begin = mstart[b_lo] & ~15;
  const int c_end   = mstart[b_hi + 1];

  for (int c0 = c_begin; c0 < c_end; c0 += 16) {
    const int  j  = c0 + l16;
    const bool jv = (j < N_ATOMS);
    const int  js = jv ? j : (N_ATOMS - 1);

    const float qx = pos[3 * js + 0];
    const float qy = pos[3 * js + 1];
    const float qz = pos[3 * js + 2];
    const float csq = qx * qx + qy * qy + qz * qz;
    const int   cb  = batch[js];

    // B fragment: lane L holds col N=l16, K = {2*half, 2*half+1}
    v2f bfrag;
    bfrag.x = (half == 0) ? qx : qz;
    bfrag.y = (half == 0) ? qy : 0.0f;

    v8f acc = {};
    // D = A(16x4) x B(4x16) : dot products pos_i . pos_j
    acc = __builtin_amdgcn_wmma_f32_16x16x4_f32(
        /*neg_a=*/false, afrag, /*neg_b=*/false, bfrag,
        /*c_mod=*/(short)0, acc, /*reuse_a=*/false, /*reuse_b=*/false);

    // lane holds D[M][N] with N=l16, M = v + 8*half
#pragma unroll
    for (int v = 0; v < 8; ++v) {
      const int M = v + 8 * half;
      float dsq = fmaf(-2.0f, acc[v], rsq[v] + csq);
      dsq = fmaxf(dsq, 0.0f);
      float d = sqrtf(dsq);
      const int  grow   = r0 + M;
      const bool masked = (!jv) | (cb != rb[v]) | (grow == j);
      d = masked ? CUTOFF_F : fminf(d, CUTOFF_F);
      s_tile[M * 16 + l16] = d;
    }
    __syncthreads();

    // per-row top-K insertion: lane m (<16) exclusively owns row m's list
    if (half == 0) {
      float* lw = s_lw[l16];
      int*   li = s_li[l16];
#pragma unroll 1
      for (int n = 0; n < 16; ++n) {
        const float d = s_tile[l16 * 16 + n];
        if (d < lw[K_NB - 1]) {
          int p = K_NB - 1;
          while (p > 0 && lw[p - 1] > d) {
            lw[p] = lw[p - 1];
            li[p] = li[p - 1];
            --p;
          }
          lw[p] = d;
          li[p] = c0 + n;
        }
      }
    }
    __syncthreads();
  }

  // emit: edge_index[0]=neighbor idx, edge_index[1]=row, then edge_weight
  int* e0 = eidx;                       // [N*K]
  int* e1 = eidx + N_ATOMS * K_NB;      // [N*K]
  for (int m = 0; m < 16; ++m) {
    const int row = r0 + m;
    const int o   = row * K_NB + lane;  // K_NB == 32 == wave width
    e0[o] = s_li[m][lane];
    e1[o] = row;
    ew[o] = s_lw[m][lane];
  }
}

// ---------------------------------------------------------------------------
extern "C" void kernel_launch(void* const* d_in, const int* in_sizes, int n_in,
                              void* d_out, int out_size, void* d_ws, size_t ws_size,
                              hipStream_t stream) {
  const float* pos   = (const float*)d_in[0];   // [N,3] f32
  const int*   batch = (const int*)d_in[1];     // [N] i32 (sorted)

  int* mstart = (int*)d_ws;                     // NUM_MOLS+1 ints

  // outputs: edge_index [2, N*K] int32, then edge_weight [N*K] f32
  int*   eidx = (int*)d_out;
  float* ew   = (float*)d_out + 2 * N_ATOMS * K_NB;

  mol_ranges_kernel<<<1, 256, 0, stream>>>(batch, mstart, N_ATOMS, NUM_MOLS);

  knn_tile_kernel<<<N_ATOMS / 16, 32, 0, stream>>>(pos, batch, mstart, eidx, ew);
}